// BaseModel_83777632076506
// MI455X (gfx1250) — compile-verified
//
#include <hip/hip_runtime.h>

// Problem constants (match reference)
#define B_ 8
#define C_ 128
#define N_ 131072
#define P_ 128
#define D_ 128
#define M_ (B_ * P_)        // 1024 columns for the projection GEMMs
#define EPS_ 1e-5f

typedef __attribute__((ext_vector_type(2))) float v2f;
typedef __attribute__((ext_vector_type(8))) float v8f;

// ---------------------------------------------------------------------------
// Utility: zero the pooling accumulators (must be re-zeroed every call).
// ---------------------------------------------------------------------------
__global__ void zero_f32(float* __restrict__ p, int n) {
  int i = blockIdx.x * blockDim.x + threadIdx.x;
  if (i < n) p[i] = 0.0f;
}

// ---------------------------------------------------------------------------
// Stage 1: segment-sum pooling.
// grid = (512 chunks, 2 channel halves), block = 256 threads.
// Each block handles one batch b, a 2048-point chunk, and 64 channels.
// Coalesced global reads of features; scatter privatized into LDS with
// ds_add_f32, flushed once per block with global f32 atomics.
// LDS: lsum[c][p] so atomic bank = label (varies per lane).
// ---------------------------------------------------------------------------
#define NCHUNK 2048
#define CH_HALF 64

__global__ void pool_accum(const float* __restrict__ feats,   // [B][C][N]
                           const int* __restrict__ labels,    // [B][N]
                           float* __restrict__ gsum,          // [B][C][P]
                           float* __restrict__ gcnt)          // [B][P]
{
  __shared__ float lsum[CH_HALF * P_];   // 32 KB
  __shared__ float lcnt[P_];
  __shared__ int   llab[NCHUNK];         // 8 KB

  const int tid   = threadIdx.x;         // 256 threads
  const int blk   = blockIdx.x;          // 0..511
  const int b     = blk >> 6;            // / 64 chunks
  const int chunk = blk & 63;
  const int n0    = chunk * NCHUNK;
  const int cBase = blockIdx.y * CH_HALF;

  for (int i = tid; i < CH_HALF * P_; i += 256) lsum[i] = 0.0f;
  if (tid < P_) lcnt[tid] = 0.0f;
  for (int i = tid; i < NCHUNK; i += 256)
    llab[i] = labels[b * N_ + n0 + i];
  __syncthreads();

  if (blockIdx.y == 0) {
    for (int i = tid; i < NCHUNK; i += 256)
      atomicAdd(&lcnt[llab[i]], 1.0f);
  }

  for (int cc = 0; cc < CH_HALF; ++cc) {
    const float* __restrict__ src =
        feats + (size_t)(b * C_ + cBase + cc) * N_ + n0;
    float* __restrict__ dst = &lsum[cc * P_];
    for (int i = tid; i < NCHUNK; i += 256)
      atomicAdd(&dst[llab[i]], src[i]);          // ds_add_f32
  }
  __syncthreads();

  for (int i = tid; i < CH_HALF * P_; i += 256) {
    int cc = i >> 7, p = i & 127;
    atomicAdd(&gsum[b * (C_ * P_) + (cBase + cc) * P_ + p], lsum[i]);
  }
  if (blockIdx.y == 0 && tid < P_)
    atomicAdd(&gcnt[b * P_ + tid], lcnt[tid]);
}

// ---------------------------------------------------------------------------
// Stage 2: pooled[c][m] = gsum[b][c][p] / max(count,1), m = b*128+p.
// Also emits the counts output (second tuple element of the reference).
// ---------------------------------------------------------------------------
__global__ void pool_finalize(const float* __restrict__ gsum,
                              const float* __restrict__ gcnt,
                              float* __restrict__ pooled,      // [C][M]
                              float* __restrict__ out_counts)  // [B*P]
{
  int idx = blockIdx.x * blockDim.x + threadIdx.x;  // 0..C*M-1
  if (idx >= C_ * M_) return;
  int c = idx >> 10;
  int m = idx & (M_ - 1);
  int b = m >> 7;
  int p = m & 127;
  float cnt = gcnt[m];
  float s   = gsum[b * (C_ * P_) + c * P_ + p];
  pooled[idx] = s / (cnt != 0.0f ? cnt : 1.0f);
  if (idx < M_) out_counts[idx] = cnt;
}

// ---------------------------------------------------------------------------
// Stage 3: 128x128 GEMM against 1024 columns using V_WMMA_F32_16X16X4_F32
// (full fp32 — matches reference arithmetic). Out[d][m] = sum_k A[d][k]*Bm[k][m].
// One wave per 16x16 tile; K=128 as 32 chained 16x16x4 accumulations.
// Per ISA layout: lanes 0-15 carry K = {k0,k0+1}, lanes 16-31 carry {k0+2,k0+3}
// for both A and B; C/D row r lives in VGPR r, lane halves = M/N split.
// ---------------------------------------------------------------------------
__global__ void gemm_wmma(const float* __restrict__ A,    // [128][128] (d,k)
                          const float* __restrict__ Bm,   // [128][1024] (k,m)
                          float* __restrict__ Out)        // [128][1024] (d,m)
{
  const int lane  = threadIdx.x & 31;
  const int wave  = threadIdx.x >> 5;
  const int tile  = blockIdx.x * 4 + wave;   // 512 tiles total
  const int colT  = tile >> 3;               // 64 column tiles
  const int dT    = tile & 7;                // 8 row tiles
  const int lm    = lane & 15;
  const int half  = lane >> 4;
  const int dBase = dT * 16;
  const int nBase = colT * 16;

  v8f acc = {};
  const float* __restrict__ aRow = A + (dBase + lm) * C_;
  const float* __restrict__ bCol = Bm + nBase + lm;

#pragma unroll
  for (int k0 = 0; k0 < C_; k0 += 4) {
    const int ka = k0 + 2 * half;
    v2f av; av.x = aRow[ka];         av.y = aRow[ka + 1];
    v2f bv; bv.x = bCol[ka * M_];    bv.y = bCol[(ka + 1) * M_];
    acc = __builtin_amdgcn_wmma_f32_16x16x4_f32(
        /*neg_a=*/false, av, /*neg_b=*/false, bv,
        /*c_mod=*/(short)0, acc, /*reuse_a=*/false, /*reuse_b=*/false);
  }

#pragma unroll
  for (int r = 0; r < 8; ++r)
    Out[(dBase + r + 8 * half) * M_ + nBase + lm] = acc[r];
}

// ---------------------------------------------------------------------------
// Stage 4: training-mode BatchNorm over the 1024 columns of one channel,
// fused scale/shift (+ optional ReLU). One block (256 thr) per channel.
// finalLayout=1 writes d_out in [B][D][P] order.
// Note: conv biases b1/b2 cancel exactly under training-mode BN (they shift
// x and its mean identically, variance unchanged) so they are never applied.
// ---------------------------------------------------------------------------
__global__ void bn_apply(const float* __restrict__ Y,     // [128][1024]
                         const float* __restrict__ g,
                         const float* __restrict__ beta,
                         float* __restrict__ Out,
                         int doRelu, int finalLayout)
{
  __shared__ float red[16];
  __shared__ float sc[2];
  const int d   = blockIdx.x;
  const int tid = threadIdx.x;  // 256 = 8 waves

  float v[4], s = 0.0f, s2 = 0.0f;
#pragma unroll
  for (int i = 0; i < 4; ++i) {
    v[i] = Y[d * M_ + tid + 256 * i];
    s  += v[i];
    s2 += v[i] * v[i];
  }
  // wave32 butterfly reduce
  for (int off = 16; off > 0; off >>= 1) {
    s  += __shfl_down(s, off, 32);
    s2 += __shfl_down(s2, off, 32);
  }
  if ((tid & 31) == 0) { red[tid >> 5] = s; red[8 + (tid >> 5)] = s2; }
  __syncthreads();
  if (tid == 0) {
    float ts = 0.0f, ts2 = 0.0f;
    for (int w = 0; w < 8; ++w) { ts += red[w]; ts2 += red[8 + w]; }
    float mean  = ts  * (1.0f / M_);
    float var   = ts2 * (1.0f / M_) - mean * mean;
    float scale = g[d] * rsqrtf(var + EPS_);
    sc[0] = scale;
    sc[1] = beta[d] - mean * scale;
  }
  __syncthreads();
  const float scale = sc[0], shift = sc[1];
#pragma unroll
  for (int i = 0; i < 4; ++i) {
    int m = tid + 256 * i;
    float val = v[i] * scale + shift;
    if (doRelu) val = fmaxf(val, 0.0f);
    if (finalLayout) {
      int b = m >> 7, p = m & 127;
      Out[b * (D_ * P_) + d * P_ + p] = val;   // [B][D][P]
    } else {
      Out[d * M_ + m] = val;
    }
  }
}

// ---------------------------------------------------------------------------
extern "C" void kernel_launch(void* const* d_in, const int* in_sizes, int n_in,
                              void* d_out, int out_size, void* d_ws, size_t ws_size,
                              hipStream_t stream) {
  const float* feats  = (const float*)d_in[0];
  const int*   labels = (const int*)  d_in[1];
  const float* W1     = (const float*)d_in[2];
  /* d_in[3] = b1: unused — cancels exactly under training-mode BN */
  const float* g1     = (const float*)d_in[4];
  const float* be1    = (const float*)d_in[5];
  const float* W2     = (const float*)d_in[6];
  /* d_in[7] = b2: unused — same cancellation */
  const float* g2     = (const float*)d_in[8];
  const float* be2    = (const float*)d_in[9];
  const float* W3     = (const float*)d_in[10];
  const float* g3     = (const float*)d_in[11];
  const float* be3    = (const float*)d_in[12];
  float* out = (float*)d_out;   // [x: 131072][counts: 1024]

  float* ws     = (float*)d_ws;
  float* gsum   = ws;               // 131072 f32  [B][C][P]
  float* gcnt   = gsum + 131072;    //   1024 f32  [B][P]   (contiguous after gsum)
  float* pooled = gcnt + 1024;      // 131072 f32  [C][M]
  float* buf1   = pooled + 131072;  // 131072 f32
  float* buf2   = buf1 + 131072;    // 131072 f32

  // Zero pooling accumulators (gsum + gcnt are contiguous: 132096 floats)
  zero_f32<<<(132096 + 255) / 256, 256, 0, stream>>>(gsum, 132096);

  // Segment-mean pooling
  pool_accum<<<dim3(512, 2), 256, 0, stream>>>(feats, labels, gsum, gcnt);
  pool_finalize<<<(C_ * M_) / 256, 256, 0, stream>>>(gsum, gcnt, pooled,
                                                     out + 131072);

  // Projection head: (GEMM -> BN -> ReLU) x2, then GEMM -> BN
  gemm_wmma<<<128, 128, 0, stream>>>(W1, pooled, buf1);
  bn_apply <<<128, 256, 0, stream>>>(buf1, g1, be1, buf2, 1, 0);
  gemm_wmma<<<128, 128, 0, stream>>>(W2, buf2, buf1);
  bn_apply <<<128, 256, 0, stream>>>(buf1, g2, be2, buf2, 1, 0);
  gemm_wmma<<<128, 128, 0, stream>>>(W3, buf2, buf1);
  bn_apply <<<128, 256, 0, stream>>>(buf1, g3, be3, out, 0, 1);
}